// LinearDrugResponsePrior_82360292868297
// MI455X (gfx1250) — compile-verified
//
#include <hip/hip_runtime.h>
#include <hip/hip_bf16.h>
#include <math.h>

#define N_RESP  524288
#define N_CELLS 1024
#define N_MISS  128
#define N_DRUGS 265
#define N_FEAT  256
#define NDOSES  9
#define BATCH   524288

typedef float v2f __attribute__((ext_vector_type(2)));
typedef float v8f __attribute__((ext_vector_type(8)));

// ---------------------------------------------------------------------------
// Kernel 1: cell_embed_raw[1024 x 9] = cell_features[1024 x 256] @ mut[256 x 9]
// via V_WMMA_F32_16X16X4_F32. One wave per 16-row M-tile, 8 waves per block,
// 8 blocks. B (256 x 16, doses zero-padded to 16) staged in LDS.
// ---------------------------------------------------------------------------
__global__ __launch_bounds__(256) void wmma_cell_embed_kernel(
    const float* __restrict__ cellF,   // [N_CELLS, N_FEAT] row-major
    const float* __restrict__ mutF,    // [N_FEAT, NDOSES] row-major
    float* __restrict__ raw)           // [N_CELLS, NDOSES]
{
    __shared__ float Bs[N_FEAT * 16];  // 16 KB, doses padded 9 -> 16

    const int t = threadIdx.x;
    for (int e = t; e < N_FEAT * 16; e += 256) {
        const int k = e >> 4;
        const int n = e & 15;
        Bs[e] = (n < NDOSES) ? mutF[k * NDOSES + n] : 0.0f;
    }
    __syncthreads();

    const int wave   = t >> 5;
    const int lane   = t & 31;
    const int tile   = blockIdx.x * 8 + wave;        // 0..63
    const int mlocal = lane & 15;                    // A row / D column index
    const int kfrag  = (lane >> 4) << 1;             // lanes>=16 own K+2,K+3
    const int row    = tile * 16 + mlocal;

    const float* __restrict__ Arow = cellF + (size_t)row * N_FEAT;

    v8f c = {};
    #pragma unroll 8
    for (int k = 0; k < N_FEAT; k += 4) {
        // A 16x4 fragment: lane holds A[m][k+kfrag], A[m][k+kfrag+1]
        const float2 av = *(const float2*)(Arow + k + kfrag);
        v2f a; a.x = av.x; a.y = av.y;
        // B 4x16 fragment: lane holds B[k+kfrag][n], B[k+kfrag+1][n]
        v2f b;
        b.x = Bs[(k + kfrag)     * 16 + mlocal];
        b.y = Bs[(k + kfrag + 1) * 16 + mlocal];
        c = __builtin_amdgcn_wmma_f32_16x16x4_f32(
                /*neg_a=*/false, a, /*neg_b=*/false, b,
                /*c_mod=*/(short)0, c, /*reuse_a=*/false, /*reuse_b=*/false);
    }

    // D layout: VGPR v -> M = v (lanes 0-15) / v+8 (lanes 16-31); N = lane&15
    if (mlocal < NDOSES) {
        const int rbase = tile * 16 + ((lane >> 4) << 3);
        #pragma unroll
        for (int v = 0; v < 8; ++v)
            raw[(size_t)(rbase + v) * NDOSES + mlocal] = c[v];
    }
}

// ---------------------------------------------------------------------------
// Kernel 2: fwd_to_mu over all 1024 + 128 + 265 table rows.
// mu[0] = f[0]; mu[j] = f[0] + sum_{t=1..j} softplus(f[t])
// ---------------------------------------------------------------------------
__device__ __forceinline__ float softplus_stable(float x) {
    return fmaxf(x, 0.0f) + log1pf(expf(-fabsf(x)));
}

__global__ __launch_bounds__(256) void fwd_to_mu_kernel(
    const float* __restrict__ raw,     // [N_CELLS, NDOSES]
    const float* __restrict__ missE,   // [N_MISS, NDOSES]
    const float* __restrict__ drugE,   // [N_DRUGS, NDOSES]
    float* __restrict__ muCell,
    float* __restrict__ muMiss,
    float* __restrict__ muDrug)
{
    const int r = blockIdx.x * 256 + threadIdx.x;
    const float* src;
    float* dst;
    if (r < N_CELLS) {
        src = raw + (size_t)r * NDOSES;            dst = muCell + (size_t)r * NDOSES;
    } else if (r < N_CELLS + N_MISS) {
        const int q = r - N_CELLS;
        src = missE + (size_t)q * NDOSES;          dst = muMiss + (size_t)q * NDOSES;
    } else if (r < N_CELLS + N_MISS + N_DRUGS) {
        const int q = r - (N_CELLS + N_MISS);
        src = drugE + (size_t)q * NDOSES;          dst = muDrug + (size_t)q * NDOSES;
    } else {
        return;
    }
    float acc = src[0];
    dst[0] = acc;
    #pragma unroll
    for (int j = 1; j < NDOSES; ++j) {
        acc += softplus_stable(src[j]);
        dst[j] = acc;
    }
}

// ---------------------------------------------------------------------------
// Kernel 3: per-response gather + add (HBM-write-bound).
// ---------------------------------------------------------------------------
__global__ __launch_bounds__(256) void gather_add_kernel(
    const int* __restrict__ idx,
    const int* __restrict__ tidx,
    const int* __restrict__ is_missing,
    const int* __restrict__ cell_map,
    const int* __restrict__ drug_map,
    const float* __restrict__ muCell,
    const float* __restrict__ muMiss,
    const float* __restrict__ muDrug,
    float* __restrict__ out)
{
    const int i = blockIdx.x * 256 + threadIdx.x;
    if (i >= BATCH) return;

    const int id0  = idx[i];
    const int id1  = tidx[i];
    const int miss = is_missing[id0];
    const int cm   = cell_map[id0];
    const int dm   = drug_map[id1];

    const float* __restrict__ crow =
        miss ? (muMiss + (size_t)cm * NDOSES) : (muCell + (size_t)cm * NDOSES);
    const float* __restrict__ drow = muDrug + (size_t)dm * NDOSES;

    float* __restrict__ o = out + (size_t)i * NDOSES;
    #pragma unroll
    for (int j = 0; j < NDOSES; ++j)
        o[j] = crow[j] + drow[j];
}

// ---------------------------------------------------------------------------
extern "C" void kernel_launch(void* const* d_in, const int* in_sizes, int n_in,
                              void* d_out, int out_size, void* d_ws, size_t ws_size,
                              hipStream_t stream) {
    const int*   idx    = (const int*)  d_in[0];
    const int*   tidx   = (const int*)  d_in[1];
    const int*   is_mis = (const int*)  d_in[2];
    const int*   cmap   = (const int*)  d_in[3];
    const int*   dmap   = (const int*)  d_in[4];
    const float* cellF  = (const float*)d_in[5];
    const float* mutF   = (const float*)d_in[6];
    const float* missE  = (const float*)d_in[7];
    const float* drugE  = (const float*)d_in[8];
    float*       out    = (float*)d_out;

    // Workspace layout (16B-aligned chunks)
    char* ws = (char*)d_ws;
    float* raw    = (float*)(ws + 0);        // 1024*9*4 = 36864 B
    float* muCell = (float*)(ws + 36864);    // 36864 B
    float* muMiss = (float*)(ws + 73728);    // 128*9*4 = 4608 B
    float* muDrug = (float*)(ws + 78336);    // 265*9*4 = 9540 B
    (void)ws_size; (void)in_sizes; (void)n_in; (void)out_size;

    // 1) WMMA precompute of the 1024x9 cell-embedding table
    wmma_cell_embed_kernel<<<8, 256, 0, stream>>>(cellF, mutF, raw);

    // 2) fwd_to_mu over all table rows (1024 + 128 + 265 = 1417)
    fwd_to_mu_kernel<<<(N_CELLS + N_MISS + N_DRUGS + 255) / 256, 256, 0, stream>>>(
        raw, missE, drugE, muCell, muMiss, muDrug);

    // 3) Main gather + add over the batch
    gather_add_kernel<<<BATCH / 256, 256, 0, stream>>>(
        idx, tidx, is_mis, cmap, dmap, muCell, muMiss, muDrug, out);
}